// AxialSelfAttentionModule_2482491097694
// MI455X (gfx1250) — compile-verified
//
#include <hip/hip_runtime.h>
#include <math.h>
#include <stdint.h>

// ---------------- problem constants ----------------
constexpr int BB  = 2;
constexpr int C   = 96;
constexpr int Dd  = 40;
constexpr int Hh  = 40;
constexpr int Ww  = 40;
constexpr int NSP = Dd * Hh * Ww;      // 64000
constexpr int NH  = 4;
constexpr int HD  = 24;                // C / NH
constexpr int L   = 40;                // axial sequence length (all dims 40)
constexpr int C3  = 3 * C;             // 288

typedef float v2f __attribute__((ext_vector_type(2)));
typedef float v8f __attribute__((ext_vector_type(8)));
typedef unsigned int u32x4 __attribute__((ext_vector_type(4)));
typedef int i32x4 __attribute__((ext_vector_type(4)));
typedef int i32x8 __attribute__((ext_vector_type(8)));

// ---------------- small device helpers ----------------
__device__ __forceinline__ float gelu_exact(float v) {
    return 0.5f * v * (1.0f + erff(v * 0.70710678118654752440f));
}

__device__ __forceinline__ void cross3(const float* a, const float* b, float* c) {
    c[0] = a[1] * b[2] - a[2] * b[1];
    c[1] = a[2] * b[0] - a[0] * b[2];
    c[2] = a[0] * b[1] - a[1] * b[0];
}

__device__ __forceinline__ void norm3(float* v) {
    float n = sqrtf(v[0] * v[0] + v[1] * v[1] + v[2] * v[2]);
    n = fmaxf(n, 1e-8f);
    v[0] /= n; v[1] /= n; v[2] /= n;
}

__device__ void keep_r6_dev(const float* r6, float* v1, float* v2) {
    v1[0] = r6[0]; v1[1] = r6[1]; v1[2] = r6[2];
    norm3(v1);
    float t[3] = {r6[3], r6[4], r6[5]};
    float d = t[0] * v1[0] + t[1] * v1[1] + t[2] * v1[2];
    t[0] -= d * v1[0]; t[1] -= d * v1[1]; t[2] -= d * v1[2];
    float n2 = sqrtf(t[0] * t[0] + t[1] * t[1] + t[2] * t[2]);
    const float e1[3] = {1.f, 0.f, 0.f};
    const float e2[3] = {0.f, 1.f, 0.f};
    const float* fb = (fabsf(v1[0]) < 0.9f) ? e1 : e2;
    float fbv[3];
    cross3(v1, fb, fbv);
    float mask = (n2 < 1e-6f) ? 1.0f : 0.0f;
    v2[0] = (1.f - mask) * t[0] + mask * fbv[0];
    v2[1] = (1.f - mask) * t[1] + mask * fbv[1];
    v2[2] = (1.f - mask) * t[2] + mask * fbv[2];
    norm3(v2);
}

// Build the 12 rotation matrices (3 axes x 4 heads) from R6 params.
__global__ void build_rot_kernel(const float* __restrict__ R6d,
                                 const float* __restrict__ R6h,
                                 const float* __restrict__ R6w,
                                 float* __restrict__ Rout) {
    int t = threadIdx.x;
    if (t >= 3 * NH) return;
    int axis = t / NH, head = t % NH;
    const float* src = (axis == 0) ? R6d : (axis == 1) ? R6h : R6w;
    const float* r6 = src + head * 6;  // shape (1, NH, 6)
    float v1[3], v2[3];
    keep_r6_dev(r6, v1, v2);           // keep_r6(R6)
    float r6b[6] = {v1[0], v1[1], v1[2], v2[0], v2[1], v2[2]};
    keep_r6_dev(r6b, v1, v2);          // r6_to_matrix applies keep_r6 again
    float v3[3];
    cross3(v1, v2, v3);
    float c23[3];
    cross3(v2, v3, c23);
    float det = v1[0] * c23[0] + v1[1] * c23[1] + v1[2] * c23[2];
    float s = (det < 0.f) ? -1.f : ((det > 0.f) ? 1.f : 0.f);
    float* R = Rout + (axis * NH + head) * 9;
    for (int i = 0; i < 3; ++i) {
        R[i * 3 + 0] = v1[i] * s;      // columns are v1, v2, v3
        R[i * 3 + 1] = v2[i] * s;
        R[i * 3 + 2] = v3[i] * s;
    }
}

// ---------------- instance norm ----------------
__global__ void instnorm_stats_kernel(const float* __restrict__ X,
                                      float* __restrict__ mu,
                                      float* __restrict__ rs) {
    __shared__ float sh1[256], sh2[256];
    const int bc = blockIdx.x;
    const int tid = threadIdx.x;
    const float* p = X + (size_t)bc * NSP;
    float s = 0.f, s2 = 0.f;
    for (int i = tid; i < NSP; i += 256) {
        float v = p[i];
        s += v;
        s2 += v * v;
    }
    sh1[tid] = s; sh2[tid] = s2;
    __syncthreads();
    for (int off = 128; off > 0; off >>= 1) {
        if (tid < off) { sh1[tid] += sh1[tid + off]; sh2[tid] += sh2[tid + off]; }
        __syncthreads();
    }
    if (tid == 0) {
        float m = sh1[0] / (float)NSP;
        float var = sh2[0] / (float)NSP - m * m;
        mu[bc] = m;
        rs[bc] = rsqrtf(var + 1e-5f);
    }
}

__global__ void in_gelu_kernel(float* __restrict__ X,
                               const float* __restrict__ mu,
                               const float* __restrict__ rs) {
    size_t i = (size_t)blockIdx.x * blockDim.x + threadIdx.x;
    if (i >= (size_t)BB * C * NSP) return;
    int bc = (int)(i / NSP);
    float v = (X[i] - mu[bc]) * rs[bc];
    X[i] = gelu_exact(v);
}

__global__ void xmod_kernel(const float* __restrict__ x,
                            const float* __restrict__ mod,
                            float* __restrict__ out) {
    size_t i = (size_t)blockIdx.x * blockDim.x + threadIdx.x;
    if (i >= (size_t)BB * C * NSP) return;
    out[i] = x[i] * mod[i];
}

// ---------------- 1x1-conv GEMM with TDM-staged activation panel ----------------
// Y[b] = W(Cout x Cin) @ X[b](Cin x Ncols) + bias.
// Block = 6 waves (192 thr) = 96 output rows; each block owns 64 columns.
// Wave 0 issues one Tensor-Data-Mover 2D tile load (Cin rows x 64 cols) into LDS,
// padded 16 DWORDs per 64-DWORD row so the LDS pitch is 80 floats (bank-conflict
// free for the half-wave K-row access pattern of the WMMA B fragment).
// Each wave computes 4 N-subtiles, reusing its A fragment x4 per K-step.
constexpr int SBP = 80;   // LDS row pitch (floats) = 64 data + 16 pad

template <int ACT>  // 0 = none, 1 = sigmoid
__global__ void __launch_bounds__(192)
gemm1x1_kernel(const float* __restrict__ Wt, const float* __restrict__ bias,
               const float* __restrict__ X, float* __restrict__ Y,
               int Cout, int Cin, int Ncols) {
    __shared__ float sB[C * SBP];      // Cin==96 rows * pitch 80
    const int tid = threadIdx.x;
    const int wid = tid >> 5;
    const int lane = tid & 31;
    const int n0 = blockIdx.x * 64;
    const int mrow0 = blockIdx.y * 96;
    const size_t bx = (size_t)blockIdx.z * Cin * Ncols;
    const size_t by = (size_t)blockIdx.z * Cout * Ncols;

    if (wid == 0) {
        // ---- TDM descriptor (CDNA5 ISA) ----
        const unsigned long long ga =
            (unsigned long long)(uintptr_t)(X + bx + n0);   // tile start
        u32x4 g0;
        g0.x = 1u;                                  // count=1 valid descriptor
        g0.y = 0u;                                  // lds_addr: sB is at LDS offset 0
        g0.z = (unsigned)(ga & 0xffffffffu);        // global_addr[31:0]
        g0.w = (unsigned)((ga >> 32) & 0x01ffffffu) // global_addr[56:32]
             | (2u << 30);                          // type = 2 ("image")
        i32x8 g1;
        g1[0] = (2 << 16)        // data_size = 4 bytes
              | (1 << 20)        // pad_enable
              | (5 << 22)        // pad_interval: every 64 DWORDs
              | (15 << 25);      // pad_amount: 16 DWORDs
        g1[1] = (int)((Ncols & 0xffff) << 16);                         // dim0[15:0]
        g1[2] = (int)(((unsigned)Ncols >> 16) | ((Cin & 0xffff) << 16)); // dim0 hi | dim1 lo
        g1[3] = (int)(((unsigned)Cin >> 16) | (64u << 16));            // dim1 hi | tile_dim0=64
        g1[4] = Cin;                                                   // tile_dim1 | tile_dim2=0
        g1[5] = Ncols;                                                 // tensor_dim0_stride lo
        g1[6] = 0;                                                     // stride hi | dim1_stride lo
        g1[7] = 0;
        const i32x4 gz4 = {0, 0, 0, 0};                                // groups 2/3 unused (2D)
        const i32x8 gz8 = {0, 0, 0, 0, 0, 0, 0, 0};
        __builtin_amdgcn_tensor_load_to_lds(g0, g1, gz4, gz4, gz8, 0);
        __builtin_amdgcn_s_wait_tensorcnt(0);
    }
    __syncthreads();

    const int nlo = lane & 15;
    const int khalf = lane >> 4;
    const int kb = khalf * 2;
    const int m = mrow0 + wid * 16 + nlo;
    v8f a0 = {}, a1 = {}, a2 = {}, a3 = {};
    for (int kk = 0; kk < Cin; kk += 4) {
        v2f af;
        af.x = Wt[(size_t)m * Cin + kk + kb];
        af.y = Wt[(size_t)m * Cin + kk + kb + 1];
        const float* r0 = &sB[(kk + kb) * SBP];
        const float* r1 = &sB[(kk + kb + 1) * SBP];
        v2f b0, b1, b2, b3;
        b0.x = r0[nlo];      b0.y = r1[nlo];
        b1.x = r0[16 + nlo]; b1.y = r1[16 + nlo];
        b2.x = r0[32 + nlo]; b2.y = r1[32 + nlo];
        b3.x = r0[48 + nlo]; b3.y = r1[48 + nlo];
        a0 = __builtin_amdgcn_wmma_f32_16x16x4_f32(false, af, false, b0, (short)0, a0, false, false);
        a1 = __builtin_amdgcn_wmma_f32_16x16x4_f32(false, af, false, b1, (short)0, a1, false, false);
        a2 = __builtin_amdgcn_wmma_f32_16x16x4_f32(false, af, false, b2, (short)0, a2, false, false);
        a3 = __builtin_amdgcn_wmma_f32_16x16x4_f32(false, af, false, b3, (short)0, a3, false, false);
    }
    for (int r = 0; r < 8; ++r) {
        int mm = mrow0 + wid * 16 + r + khalf * 8;
        float bia = bias[mm];
        float v0 = a0[r] + bia, v1 = a1[r] + bia, v2 = a2[r] + bia, v3 = a3[r] + bia;
        if (ACT == 1) {
            v0 = 1.f / (1.f + __expf(-v0));
            v1 = 1.f / (1.f + __expf(-v1));
            v2 = 1.f / (1.f + __expf(-v2));
            v3 = 1.f / (1.f + __expf(-v3));
        }
        float* yp = Y + by + (size_t)mm * Ncols + n0 + nlo;
        yp[0] = v0; yp[16] = v1; yp[32] = v2; yp[48] = v3;
    }
}

// ---------------- circular 3x3x3 conv as 27-tap GEMM ----------------
// W layout: (Cout=96, Cin=96, 3,3,3); tap t = kd*9+kh*3+kw, source z' = (z+kd-1) mod D.
__global__ void __launch_bounds__(32)
conv3_gemm_kernel(const float* __restrict__ Wt, const float* __restrict__ bias,
                  const float* __restrict__ X, float* __restrict__ Y) {
    const int lane = threadIdx.x;
    const int nlo = lane & 15;
    const int khalf = lane >> 4;
    const int kb = khalf * 2;
    const int n = blockIdx.x * 16 + nlo;   // spatial index (NSP multiple of 16)
    const int m0 = blockIdx.y * 16;
    const int m = m0 + nlo;
    const int b = blockIdx.z;
    const int z = n / (Hh * Ww);
    const int rem = n % (Hh * Ww);
    const int y = rem / Ww;
    const int x = rem % Ww;
    const float* Xb = X + (size_t)b * C * NSP;
    v8f cf = {};
    for (int t = 0; t < 27; ++t) {
        int dz = t / 9 - 1, dy = (t / 3) % 3 - 1, dx = t % 3 - 1;
        int zz = z + dz; zz += (zz < 0) ? Dd : 0; zz -= (zz >= Dd) ? Dd : 0;
        int yy = y + dy; yy += (yy < 0) ? Hh : 0; yy -= (yy >= Hh) ? Hh : 0;
        int xx = x + dx; xx += (xx < 0) ? Ww : 0; xx -= (xx >= Ww) ? Ww : 0;
        const int sidx = (zz * Hh + yy) * Ww + xx;
        for (int kk = 0; kk < C; kk += 4) {
            v2f af, bf;
            af.x = Wt[((size_t)m * C + kk + kb) * 27 + t];
            af.y = Wt[((size_t)m * C + kk + kb + 1) * 27 + t];
            bf.x = Xb[(size_t)(kk + kb) * NSP + sidx];
            bf.y = Xb[(size_t)(kk + kb + 1) * NSP + sidx];
            cf = __builtin_amdgcn_wmma_f32_16x16x4_f32(false, af, false, bf,
                                                       (short)0, cf, false, false);
        }
    }
    for (int r = 0; r < 8; ++r) {
        int mm = m0 + r + khalf * 8;
        Y[((size_t)b * C + mm) * NSP + n] = cf[r] + bias[mm];
    }
}

// ---------------- axial attention (one wave per (b, a1, a2, head)) ----------------
// NOTE: pos_attn is added as a per-query constant along the softmax axis -> no-op; skipped.
template <int AXIS>
__global__ void __launch_bounds__(32)
attn_axis_kernel(const float* __restrict__ qkv, const float* __restrict__ Rmat,
                 float* __restrict__ acc, int accumulate) {
    __shared__ float sQ[L * HD];
    __shared__ float sK[L * HD];
    __shared__ float sV[L * HD];
    __shared__ float sS[L * L];

    const int lane = threadIdx.x;
    const int ap = blockIdx.x;
    const int a1 = ap / 40, a2 = ap % 40;
    const int head = blockIdx.y;
    const int b = blockIdx.z;

    int base, strideL;
    if (AXIS == 0)      { base = a1 * Ww + a2;          strideL = Hh * Ww; }  // seq over z
    else if (AXIS == 1) { base = a1 * Hh * Ww + a2;     strideL = Ww;      }  // seq over y
    else                { base = (a1 * Hh + a2) * Ww;   strideL = 1;       }  // seq over x

    const float* Rp = Rmat + (AXIS * NH + head) * 9;
    const float R00 = Rp[0], R01 = Rp[1], R02 = Rp[2];
    const float R10 = Rp[3], R11 = Rp[4], R12 = Rp[5];
    const float R20 = Rp[6], R21 = Rp[7], R22 = Rp[8];

    const size_t qoff = ((size_t)b * C3 + head * HD) * NSP;
    const size_t koff = qoff + (size_t)C * NSP;
    const size_t voff = qoff + (size_t)(2 * C) * NSP;

    // load + rotate Q and K (HD = 8 vec3 groups per position)
    for (int g = lane; g < L * 8; g += 32) {
        int l = g >> 3, e = (g & 7) * 3;
        size_t n = (size_t)(base + l * strideL);
        float q0 = qkv[qoff + (size_t)(e + 0) * NSP + n];
        float q1 = qkv[qoff + (size_t)(e + 1) * NSP + n];
        float q2 = qkv[qoff + (size_t)(e + 2) * NSP + n];
        sQ[l * HD + e + 0] = R00 * q0 + R01 * q1 + R02 * q2;
        sQ[l * HD + e + 1] = R10 * q0 + R11 * q1 + R12 * q2;
        sQ[l * HD + e + 2] = R20 * q0 + R21 * q1 + R22 * q2;
        float k0 = qkv[koff + (size_t)(e + 0) * NSP + n];
        float k1 = qkv[koff + (size_t)(e + 1) * NSP + n];
        float k2 = qkv[koff + (size_t)(e + 2) * NSP + n];
        sK[l * HD + e + 0] = R00 * k0 + R01 * k1 + R02 * k2;
        sK[l * HD + e + 1] = R10 * k0 + R11 * k1 + R12 * k2;
        sK[l * HD + e + 2] = R20 * k0 + R21 * k1 + R22 * k2;
    }
    for (int i = lane; i < L * HD; i += 32) {
        int l = i / HD, e = i % HD;
        sV[i] = qkv[voff + (size_t)e * NSP + base + l * strideL];
    }
    __syncthreads();

    const int nlo = lane & 15;
    const int khalf = lane >> 4;
    const int kb = khalf * 2;
    const float scale = 0.20412414523193150818f;  // 24^-0.5

    // S = Qr @ Kr^T * scale   (40x40, K=24)
    for (int mi = 0; mi < 3; ++mi) {
        for (int ni = 0; ni < 3; ++ni) {
            v8f cf = {};
            const int mA = mi * 16 + nlo;
            const int nB = ni * 16 + nlo;
            for (int kk = 0; kk < HD; kk += 4) {
                v2f af, bf;
                af.x = (mA < L) ? sQ[mA * HD + kk + kb]     : 0.f;
                af.y = (mA < L) ? sQ[mA * HD + kk + kb + 1] : 0.f;
                bf.x = (nB < L) ? sK[nB * HD + kk + kb]     : 0.f;  // B[k][n] = Kr[n][k]
                bf.y = (nB < L) ? sK[nB * HD + kk + kb + 1] : 0.f;
                cf = __builtin_amdgcn_wmma_f32_16x16x4_f32(false, af, false, bf,
                                                           (short)0, cf, false, false);
            }
            for (int r = 0; r < 8; ++r) {
                int mm = mi * 16 + r + khalf * 8;
                int nn = ni * 16 + nlo;
                if (mm < L && nn < L) sS[mm * L + nn] = cf[r] * scale;
            }
        }
    }
    __syncthreads();

    // row softmax (pos_attn bias is row-constant -> omitted)
    for (int row = lane; row < L; row += 32) {
        float mx = -3.4e38f;
        for (int j = 0; j < L; ++j) mx = fmaxf(mx, sS[row * L + j]);
        float s = 0.f;
        for (int j = 0; j < L; ++j) {
            float e = __expf(sS[row * L + j] - mx);
            sS[row * L + j] = e;
            s += e;
        }
        float inv = 1.f / s;
        for (int j = 0; j < L; ++j) sS[row * L + j] *= inv;
    }
    __syncthreads();

    // O = P @ V   (40x24, K=40)
    for (int mi = 0; mi < 3; ++mi) {
        for (int ni = 0; ni < 2; ++ni) {
            v8f cf = {};
            const int mA = mi * 16 + nlo;
            const int nB = ni * 16 + nlo;
            for (int kk = 0; kk < L; kk += 4) {
                v2f af, bf;
                af.x = (mA < L) ? sS[mA * L + kk + kb]     : 0.f;
                af.y = (mA < L) ? sS[mA * L + kk + kb + 1] : 0.f;
                bf.x = (nB < HD) ? sV[(kk + kb) * HD + nB]     : 0.f;
                bf.y = (nB < HD) ? sV[(kk + kb + 1) * HD + nB] : 0.f;
                cf = __builtin_amdgcn_wmma_f32_16x16x4_f32(false, af, false, bf,
                                                           (short)0, cf, false, false);
            }
            for (int r = 0; r < 8; ++r) {
                int mm = mi * 16 + r + khalf * 8;
                int hd = ni * 16 + nlo;
                if (mm < L && hd < HD) {
                    size_t idx = ((size_t)b * C + head * HD + hd) * NSP
                               + base + mm * strideL;
                    if (accumulate) acc[idx] += cf[r];
                    else            acc[idx] = cf[r];
                }
            }
        }
    }
}

// ---------------- launcher ----------------
extern "C" void kernel_launch(void* const* d_in, const int* in_sizes, int n_in,
                              void* d_out, int out_size, void* d_ws, size_t ws_size,
                              hipStream_t stream) {
    (void)in_sizes; (void)n_in; (void)out_size; (void)ws_size;
    const float* x      = (const float*)d_in[0];
    const float* pos    = (const float*)d_in[1];
    const float* qkv_w  = (const float*)d_in[2];
    const float* qkv_b  = (const float*)d_in[3];
    const float* lp1_w  = (const float*)d_in[4];
    const float* lp1_b  = (const float*)d_in[5];
    const float* lp2_w  = (const float*)d_in[6];
    const float* lp2_b  = (const float*)d_in[7];
    const float* mod1_w = (const float*)d_in[8];
    const float* mod1_b = (const float*)d_in[9];
    const float* mod2_w = (const float*)d_in[10];
    const float* mod2_b = (const float*)d_in[11];
    // d_in[12] = pa_w, d_in[13] = pa_b : provably no effect (softmax row-invariance)
    const float* R6d    = (const float*)d_in[14];
    const float* R6h    = (const float*)d_in[15];
    const float* R6w    = (const float*)d_in[16];
    const float* proj_w = (const float*)d_in[17];
    const float* proj_b = (const float*)d_in[18];
    float* out = (float*)d_out;

    const size_t NBC = (size_t)BB * C * NSP;        // 12,288,000
    float* ws   = (float*)d_ws;
    float* bufA = ws;                               // (B, C, NSP)
    float* bufB = ws + NBC;                         // (B, C, NSP)
    float* qkvb = ws + 2 * NBC;                     // (B, 3C, NSP)
    float* mu   = ws + 2 * NBC + 3 * NBC;           // 192
    float* rs   = mu + BB * C;                      // 192
    float* Rm   = rs + BB * C;                      // 108

    const dim3 w32(32);
    const dim3 w192(192);
    const dim3 gemmGrid(NSP / 64, 1, BB);           // (1000, 1, 2): 96 rows per block
    const dim3 qkvGrid(NSP / 64, C3 / 96, BB);      // (1000, 3, 2)
    const dim3 conv3Grid(NSP / 16, C / 16, BB);     // (4000, 6, 2)
    const dim3 ewGrid((unsigned)((NBC + 255) / 256));
    const dim3 attnGrid(40 * 40, NH, BB);

    // gating path
    conv3_gemm_kernel<<<conv3Grid, w32, 0, stream>>>(lp1_w, lp1_b, pos, bufA);
    instnorm_stats_kernel<<<BB * C, 256, 0, stream>>>(bufA, mu, rs);
    in_gelu_kernel<<<ewGrid, 256, 0, stream>>>(bufA, mu, rs);
    gemm1x1_kernel<0><<<gemmGrid, w192, 0, stream>>>(lp2_w, lp2_b, bufA, bufB, C, C, NSP);
    gemm1x1_kernel<0><<<gemmGrid, w192, 0, stream>>>(mod1_w, mod1_b, bufB, bufA, C, C, NSP);
    instnorm_stats_kernel<<<BB * C, 256, 0, stream>>>(bufA, mu, rs);
    in_gelu_kernel<<<ewGrid, 256, 0, stream>>>(bufA, mu, rs);
    gemm1x1_kernel<1><<<gemmGrid, w192, 0, stream>>>(mod2_w, mod2_b, bufA, bufB, C, C, NSP);
    xmod_kernel<<<ewGrid, 256, 0, stream>>>(x, bufB, bufA);   // bufA = x_mod

    // shared QKV projection (identical for all three axes)
    gemm1x1_kernel<0><<<qkvGrid, w192, 0, stream>>>(qkv_w, qkv_b, bufA, qkvb, C3, C, NSP);

    // rotation matrices, then axial attention accumulated into bufB
    build_rot_kernel<<<1, 32, 0, stream>>>(R6d, R6h, R6w, Rm);
    attn_axis_kernel<0><<<attnGrid, w32, 0, stream>>>(qkvb, Rm, bufB, 0);
    attn_axis_kernel<1><<<attnGrid, w32, 0, stream>>>(qkvb, Rm, bufB, 1);
    attn_axis_kernel<2><<<attnGrid, w32, 0, stream>>>(qkvb, Rm, bufB, 1);

    // final projection
    gemm1x1_kernel<0><<<gemmGrid, w192, 0, stream>>>(proj_w, proj_b, bufB, out, C, C, NSP);
}